// DNANet_41979010351137
// MI455X (gfx1250) — compile-verified
//
#include <hip/hip_runtime.h>
#include <hip/hip_bf16.h>

// ---------------------------------------------------------------------------
// DNA-GNN for MI455X (gfx1250).
//  * Dense projections use V_WMMA_F32_16X16X4_F32 (exact f32, matrix pipe).
//  * Edge attention phase is L2-resident gather + f32 atomic scatter
//    (working set ~150MB < 192MB L2), which is the real bottleneck; the
//    GEMMs (~6 GFLOP total) are negligible, so no reason to quantize them.
// ---------------------------------------------------------------------------

typedef __attribute__((ext_vector_type(2))) float v2f;
typedef __attribute__((ext_vector_type(8))) float v8f;

#define HID    64
#define IN_C   128
#define HEADS  4
#define DH     16

// ---------------- utility kernels ----------------
__global__ void k_fill(float* __restrict__ p, float val, int n) {
    int i = blockIdx.x * blockDim.x + threadIdx.x;
    if (i < n) p[i] = val;
}

__global__ void k_deg(const int* __restrict__ dst, float* __restrict__ deg, int E) {
    int i = blockIdx.x * blockDim.x + threadIdx.x;
    if (i < E) atomicAdd(&deg[dst[i]], 1.0f);
}

__global__ void k_rsqrt(const float* __restrict__ deg, float* __restrict__ dinv, int n) {
    int i = blockIdx.x * blockDim.x + threadIdx.x;
    if (i < n) dinv[i] = rsqrtf(deg[i]);   // deg >= 1 always (self loops)
}

// ---------------- WMMA f32 GEMM:  C[M,N] = A[M,K] @ W[K,N] + bias[N] --------
// One wave32 per 16x16 tile of C.  ISA 7.12.2 layouts:
//   A 16x4 : lane l16 = row; half 0 -> K={k,k+1}, half 1 -> K={k+2,k+3}
//   B 4x16 : lane l16 = col; half 0 -> K rows {k,k+1}, half 1 -> {k+2,k+3}
//   D 16x16: element r -> row (half*8 + r), lane l16 = col
__global__ void k_gemm_wmma(const float* __restrict__ A,
                            const float* __restrict__ W,
                            const float* __restrict__ bias,
                            float* __restrict__ C,
                            int M, int K, int N) {
    const int lane = threadIdx.x;          // 0..31
    const int half = lane >> 4;            // 0/1
    const int l16  = lane & 15;
    const int col  = blockIdx.y * 16 + l16;

    int arow = blockIdx.x * 16 + l16;
    if (arow >= M) arow = M - 1;           // clamp (keeps wave fully active)

    v8f acc = {};
    for (int k = 0; k < K; k += 4) {
        const int ka = k + half * 2;       // even -> 8B aligned
        const float2 af = *(const float2*)(A + (size_t)arow * K + ka);
        v2f a; a.x = af.x;                a.y = af.y;
        v2f b; b.x = W[(size_t)ka * N + col]; b.y = W[(size_t)(ka + 1) * N + col];
        acc = __builtin_amdgcn_wmma_f32_16x16x4_f32(
                  /*neg_a=*/false, a, /*neg_b=*/false, b,
                  /*c_mod=*/(short)0, acc, /*reuse_a=*/false, /*reuse_b=*/false);
    }

    const float bv = bias[col];
#pragma unroll
    for (int r = 0; r < 8; ++r) {
        const int m = blockIdx.x * 16 + half * 8 + r;
        if (m < M) C[(size_t)m * N + col] = acc[r] + bv;
    }
}

// ---------------- per-edge multi-head attention + scatter -------------------
// 64 threads per edge: thread c owns channel c (head = c/16).
// Per-head dot products reduced over 16 consecutive lanes via shfl_xor
// (stays inside one wave32). Softmax over L<=3 slices is register-local.
__global__ void k_edge(const int* __restrict__ ei, int E, int N,
                       const float* __restrict__ q,
                       const float* __restrict__ kb,   // [3][N][64] slice-major
                       const float* __restrict__ vb,   // [3][N][64]
                       const float* __restrict__ dinv,
                       float* __restrict__ outp,       // [N][64], pre-zeroed
                       int L) {
    const int t   = blockIdx.x * blockDim.x + threadIdx.x;
    const int eid = t >> 6;
    const int c   = t & 63;
    const int Etot = E + N;                 // real edges + self loops
    if (eid >= Etot) return;

    int s, d;
    if (eid < E) { s = ei[eid]; d = ei[E + eid]; }   // edge_index is (2,E)
    else         { s = d = eid - E; }                 // self loop

    const float  qc    = q[(size_t)d * HID + c];
    const size_t sbase = (size_t)s * HID + c;
    const size_t slice = (size_t)N * HID;

    float sc[3];
    for (int l = 0; l < L; ++l) {
        float p = qc * kb[l * slice + sbase];
        p += __shfl_xor(p, 1);
        p += __shfl_xor(p, 2);
        p += __shfl_xor(p, 4);
        p += __shfl_xor(p, 8);              // sum over the 16 lanes of a head
        sc[l] = p * 0.25f;                  // / sqrt(d=16)
    }

    float mx = sc[0];
    for (int l = 1; l < L; ++l) mx = fmaxf(mx, sc[l]);
    float w[3], den = 0.f;
    for (int l = 0; l < L; ++l) { w[l] = __expf(sc[l] - mx); den += w[l]; }
    const float inv = 1.0f / den;

    float msg = 0.f;
    for (int l = 0; l < L; ++l) msg += w[l] * vb[l * slice + sbase];
    msg *= inv * dinv[s] * dinv[d];

    atomicAdd(&outp[(size_t)d * HID + c], msg);
}

// ---------------------------------------------------------------------------
extern "C" void kernel_launch(void* const* d_in, const int* in_sizes, int n_in,
                              void* d_out, int out_size, void* d_ws, size_t ws_size,
                              hipStream_t stream) {
    const float* x     = (const float*)d_in[0];
    const int*   ei    = (const int*)  d_in[1];   // (2, E) indices
    const float* W_lin = (const float*)d_in[2];
    const float* b_lin = (const float*)d_in[3];
    const float* Wq    = (const float*)d_in[4];
    const float* bq    = (const float*)d_in[5];
    const float* Wk    = (const float*)d_in[6];
    const float* bk    = (const float*)d_in[7];
    const float* Wv    = (const float*)d_in[8];
    const float* bv    = (const float*)d_in[9];
    const float* W_out = (const float*)d_in[10];
    const float* b_out = (const float*)d_in[11];

    const int N = in_sizes[0] / IN_C;     // 50000
    const int E = in_sizes[1] / 2;        // 600000

    // Workspace carve-up (floats): deg, dinv, xs0..xs3, q, kb[3], vb[3]
    float*  ws  = (float*)d_ws;
    size_t  off = 0;
    float* deg  = ws + off; off += (size_t)N;
    float* dinv = ws + off; off += (size_t)N;
    float* xs0  = ws + off; off += (size_t)N * HID;   // h
    float* xs1  = ws + off; off += (size_t)N * HID;
    float* xs2  = ws + off; off += (size_t)N * HID;
    float* xs3  = ws + off; off += (size_t)N * HID;
    float* qb   = ws + off; off += (size_t)N * HID;
    float* kb   = ws + off; off += (size_t)3 * N * HID;
    float* vb   = ws + off; off += (size_t)3 * N * HID;

    // ---- degree / norm ----
    k_fill <<<(N + 255) / 256, 256, 0, stream>>>(deg, 1.0f, N);        // self loop
    k_deg  <<<(E + 255) / 256, 256, 0, stream>>>(ei + E, deg, E);      // dst row
    k_rsqrt<<<(N + 255) / 256, 256, 0, stream>>>(deg, dinv, N);

    // ---- h = x @ W_lin + b_lin ----
    const dim3 gH((N + 15) / 16, HID / 16);
    k_gemm_wmma<<<gH, 32, 0, stream>>>(x, W_lin, b_lin, xs0, N, IN_C, HID);

    float* xs[4] = {xs0, xs1, xs2, xs3};
    const int   Etot = E + N;
    const int   egrid = (Etot * 64 + 255) / 256;
    const size_t slice = (size_t)N * HID;

    for (int layer = 0; layer < 3; ++layer) {
        const float* xin = xs[layer];              // newest slice == q input
        k_gemm_wmma<<<gH, 32, 0, stream>>>(xin, Wq, bq, qb,               N, HID, HID);
        k_gemm_wmma<<<gH, 32, 0, stream>>>(xin, Wk, bk, kb + layer*slice, N, HID, HID);
        k_gemm_wmma<<<gH, 32, 0, stream>>>(xin, Wv, bv, vb + layer*slice, N, HID, HID);

        k_fill<<<((int)slice + 255) / 256, 256, 0, stream>>>(xs[layer + 1], 0.0f, (int)slice);
        k_edge<<<egrid, 256, 0, stream>>>(ei, E, N, qb, kb, vb, dinv,
                                          xs[layer + 1], layer + 1);
    }

    // ---- y = xs[-1] @ W_out + b_out ----
    const dim3 gO((N + 15) / 16, 1);
    k_gemm_wmma<<<gO, 32, 0, stream>>>(xs3, W_out, b_out, (float*)d_out, N, HID, 16);
}